// HPC_Group_88433376624926
// MI455X (gfx1250) — compile-verified
//
#include <hip/hip_runtime.h>

#define B_  8
#define N_  8192
#define P_  2048
#define S_  32
#define C_  64
#define R2_ 0.01f
#define NCH 70  // 3 rel + 3 gt + 64 feat

typedef __attribute__((ext_vector_type(2))) float v2f;
typedef __attribute__((ext_vector_type(8))) float v8f;

// ---------------------------------------------------------------------------
// Phase 1: ball query. One wave32 handles 16 centroids; distance matrix built
// 16x16 at a time with V_WMMA_F32_16X16X4_F32 using the K=4 trick:
//   A row m  = (-2qx, -2qy, -2qz, 1)
//   B col n  = ( xx ,  xy ,  xz , |x|^2)
//   tile     = |x|^2 - 2 q.x   ;  d2 < r2  <=>  tile < r2 - |q|^2 (per row)
// ---------------------------------------------------------------------------
__global__ __launch_bounds__(256) void HPC_ball_query_wmma(
    const float* __restrict__ xyz, const float* __restrict__ new_xyz,
    int* __restrict__ idxOut)
{
  const int lane = threadIdx.x & 31;
  const int wave = blockIdx.x * (blockDim.x >> 5) + (threadIdx.x >> 5);
  const int b    = wave / (P_ / 16);
  const int p0   = (wave % (P_ / 16)) * 16;
  const int m    = lane & 15;   // row (centroid) / col (point) within tile
  const int half = lane >> 4;   // selects K=0,1 vs K=2,3 halves of A and B

  const float* xyzb = xyz + (size_t)b * N_ * 3;
  const float* qp   = new_xyz + ((size_t)b * P_ + (p0 + m)) * 3;
  const float qx = qp[0], qy = qp[1], qz = qp[2];
  const float qq = qx * qx + qy * qy + qz * qz;

  v2f a;
  a.x = half ? (-2.0f * qz) : (-2.0f * qx);   // K=2 : K=0
  a.y = half ? 1.0f         : (-2.0f * qy);   // K=3 : K=1

  // Per-C-VGPR thresholds: C vgpr v holds row v (lanes 0-15) / v+8 (lanes 16-31)
  float thr[8];
#pragma unroll
  for (int v = 0; v < 8; ++v)
    thr[v] = R2_ - __shfl(qq, v + 8 * half, 32);

  int  cnt   = 0;
  int  first = -1;
  int* myOut = idxOut + ((size_t)(b * P_ + p0 + m)) * S_;

  for (int n0 = 0; n0 < N_; n0 += 16) {
    const int n = n0 + m;
    const float* pp = xyzb + (size_t)n * 3;
    const float px = pp[0], py = pp[1], pz = pp[2];
    if (n0 + 16 < N_)
      __builtin_prefetch(xyzb + (size_t)(n + 16) * 3, 0, 1);  // global_prefetch_b8

    v2f bb;
    bb.x = half ? pz : px;                                // K=2 : K=0
    bb.y = half ? (px * px + py * py + pz * pz) : py;     // K=3 : K=1

    v8f c = {0.f, 0.f, 0.f, 0.f, 0.f, 0.f, 0.f, 0.f};
    c = __builtin_amdgcn_wmma_f32_16x16x4_f32(
            /*neg_a=*/false, a, /*neg_b=*/false, bb,
            /*c_mod=*/(short)0, c, /*reuse_a=*/false, /*reuse_b=*/false);

    unsigned bal0 = __builtin_amdgcn_ballot_w32(c[0] < thr[0]);
    unsigned bal1 = __builtin_amdgcn_ballot_w32(c[1] < thr[1]);
    unsigned bal2 = __builtin_amdgcn_ballot_w32(c[2] < thr[2]);
    unsigned bal3 = __builtin_amdgcn_ballot_w32(c[3] < thr[3]);
    unsigned bal4 = __builtin_amdgcn_ballot_w32(c[4] < thr[4]);
    unsigned bal5 = __builtin_amdgcn_ballot_w32(c[5] < thr[5]);
    unsigned bal6 = __builtin_amdgcn_ballot_w32(c[6] < thr[6]);
    unsigned bal7 = __builtin_amdgcn_ballot_w32(c[7] < thr[7]);

    // Lane m extracts the 16-bit mask of row m (cndmask tree, no scratch).
    unsigned x0 = (m & 1) ? bal1 : bal0;
    unsigned x1 = (m & 1) ? bal3 : bal2;
    unsigned x2 = (m & 1) ? bal5 : bal4;
    unsigned x3 = (m & 1) ? bal7 : bal6;
    unsigned y0 = (m & 2) ? x1 : x0;
    unsigned y1 = (m & 2) ? x3 : x2;
    unsigned z  = (m & 4) ? y1 : y0;
    unsigned rowmask = (m & 8) ? (z >> 16) : (z & 0xFFFFu);

    if (lane < 16) {                 // lanes 16-31 would duplicate rows 0-15
      while (rowmask && cnt < S_) {  // ordered: lowest point index first
        const int bit  = __builtin_ctz(rowmask);
        const int nidx = n0 + bit;
        if (first < 0) first = nidx;
        myOut[cnt] = nidx;
        ++cnt;
        rowmask &= rowmask - 1;
      }
    }

    // Early exit when all 16 centroids of this wave are full (full-EXEC ballot).
    const bool done = (lane >= 16) || (cnt >= S_);
    if (__builtin_amdgcn_ballot_w32(done) == 0xFFFFFFFFu) break;
  }

  if (lane < 16) {
    const int pad = (first < 0) ? 0 : first;
    for (int s = cnt; s < S_; ++s) myOut[s] = pad;
  }
}

// ---------------------------------------------------------------------------
// Phase 2: gather + scatter (memory-bound, ~147 MB of stores).
// One wave per centroid; lane = sample s. Coalesced 128 B store per channel.
// ---------------------------------------------------------------------------
__global__ __launch_bounds__(256) void HPC_group_gather(
    const float* __restrict__ xyz, const float* __restrict__ new_xyz,
    const float* __restrict__ feat, const int* __restrict__ idxIn,
    float* __restrict__ out)
{
  const int lane = threadIdx.x & 31;
  const int wave = blockIdx.x * (blockDim.x >> 5) + (threadIdx.x >> 5);
  const int b = wave / P_;
  const int p = wave % P_;

  const int idx = idxIn[((size_t)(b * P_ + p)) * S_ + lane];

  const float* pp = xyz + ((size_t)b * N_ + idx) * 3;
  const float* qp = new_xyz + ((size_t)b * P_ + p) * 3;
  const float rx = pp[0] - qp[0];
  const float ry = pp[1] - qp[1];
  const float rz = pp[2] - qp[2];

  const size_t cs    = (size_t)P_ * S_;                       // channel stride
  const size_t obase = (((size_t)b * NCH) * P_ + p) * S_ + lane;
  out[obase + 0 * cs] = rx;
  out[obase + 1 * cs] = ry;
  out[obase + 2 * cs] = rz;

  // Wave32 mean of relative offsets -> gt feature (broadcast over samples).
  float sx = rx, sy = ry, sz = rz;
#pragma unroll
  for (int o = 16; o > 0; o >>= 1) {
    sx += __shfl_xor(sx, o, 32);
    sy += __shfl_xor(sy, o, 32);
    sz += __shfl_xor(sz, o, 32);
  }
  const float inv = 1.0f / ((float)S_ * 0.1f);  // mean / RADIUS
  out[obase + 3 * cs] = sx * inv;
  out[obase + 4 * cs] = sy * inv;
  out[obase + 5 * cs] = sz * inv;

  const float* fb = feat + ((size_t)b * C_) * N_;
#pragma unroll 4
  for (int c = 0; c < C_; ++c)
    out[obase + (size_t)(6 + c) * cs] = fb[(size_t)c * N_ + idx];
}

// ---------------------------------------------------------------------------
extern "C" void kernel_launch(void* const* d_in, const int* in_sizes, int n_in,
                              void* d_out, int out_size, void* d_ws, size_t ws_size,
                              hipStream_t stream) {
  const float* xyz     = (const float*)d_in[0];  // (8, 8192, 3)
  const float* new_xyz = (const float*)d_in[1];  // (8, 2048, 3)
  const float* feat    = (const float*)d_in[2];  // (8, 64, 8192)
  float* out = (float*)d_out;                    // (8, 70, 2048, 32)
  int* idxbuf = (int*)d_ws;                      // 8*2048*32 int32 = 2 MB

  // Phase 1: 8 * (2048/16) = 1024 waves, 8 waves (256 thr) per block.
  HPC_ball_query_wmma<<<dim3((B_ * (P_ / 16)) / 8), dim3(256), 0, stream>>>(
      xyz, new_xyz, idxbuf);

  // Phase 2: one wave per (b,p) -> 16384 waves -> 2048 blocks of 256.
  HPC_group_gather<<<dim3((B_ * P_) / 8), dim3(256), 0, stream>>>(
      xyz, new_xyz, feat, idxbuf, out);
}